// CayleyConv_22978075034228
// MI455X (gfx1250) — compile-verified
//
#include <hip/hip_runtime.h>
#include <math.h>

#if defined(__gfx1250__)
#if !__has_builtin(__builtin_amdgcn_wmma_f32_16x16x4_f32)
#error "missing __builtin_amdgcn_wmma_f32_16x16x4_f32 on gfx1250"
#endif
#endif

typedef __attribute__((ext_vector_type(2))) float v2f;
typedef __attribute__((ext_vector_type(8))) float v8f;

#define PI_F 3.14159265358979323846f

__device__ __forceinline__ float2 cmul(float2 a, float2 b) {
    return make_float2(a.x * b.x - a.y * b.y, a.x * b.y + a.y * b.x);
}

// ---------------------------------------------------------------------------
// K1: row-wise rfft (length 64 -> 33 bins) per (b,c) image. Direct DFT via
// 64-entry twiddle table in LDS (table index (v*n) & 63).
// out layout: [(b*128+c)*2112 + row*33 + v]  (complex)
// ---------------------------------------------------------------------------
__global__ __launch_bounds__(256) void rfft_rows_kernel(
    const float* __restrict__ x, float2* __restrict__ out) {
    __shared__ float img[64 * 64];
    __shared__ float ct[64], st[64];
    const int blk = blockIdx.x, t = threadIdx.x;
    const float* xp = x + (size_t)blk * 4096;
    for (int e = t; e < 4096; e += 256) img[e] = xp[e];
    if (t < 64) {
        float ang = (2.0f * PI_F / 64.0f) * (float)t;
        ct[t] = cosf(ang);
        st[t] = sinf(ang);
    }
    __syncthreads();
    for (int o = t; o < 2112; o += 256) {
        int row = o / 33, v = o - row * 33;
        const float* rp = img + row * 64;
        float re = 0.0f, im = 0.0f;
        for (int n = 0; n < 64; ++n) {
            int idx = (v * n) & 63;          // e^{-i 2pi v n / 64}
            float val = rp[n];
            re += val * ct[idx];
            im -= val * st[idx];
        }
        out[(size_t)blk * 2112 + o] = make_float2(re, im);
    }
}

// ---------------------------------------------------------------------------
// K2: column FFT (length-64 complex DFT) + scatter into frequency-major
// xfft layout: [(u*33+v)*4096 + c*32 + b]  (complex)
// ---------------------------------------------------------------------------
__global__ __launch_bounds__(256) void fft_cols_kernel(
    const float2* __restrict__ in, float2* __restrict__ xfft) {
    __shared__ float2 Xl[64 * 33];
    __shared__ float ct[64], st[64];
    const int blk = blockIdx.x, t = threadIdx.x;
    const int b = blk >> 7, c = blk & 127;
    const float2* ip = in + (size_t)blk * 2112;
    for (int e = t; e < 2112; e += 256) Xl[e] = ip[e];
    if (t < 64) {
        float ang = (2.0f * PI_F / 64.0f) * (float)t;
        ct[t] = cosf(ang);
        st[t] = sinf(ang);
    }
    __syncthreads();
    for (int o = t; o < 2112; o += 256) {
        int u = o / 33, v = o - u * 33;
        float re = 0.0f, im = 0.0f;
        for (int n = 0; n < 64; ++n) {
            int idx = (u * n) & 63;          // e^{-i 2pi u n / 64} = cc - i ss
            float2 s = Xl[n * 33 + v];
            float cc = ct[idx], ss = st[idx];
            re += s.x * cc + s.y * ss;
            im += s.y * cc - s.x * ss;
        }
        xfft[(size_t)o * 4096 + c * 32 + b] = make_float2(re, im);
    }
}

// ---------------------------------------------------------------------------
// K3: wfft[f][co][ci] = sm[f] * conj( sum_{p,q} w[co][ci][p][q] e^{-i2pi(up+vq)/64} )
// sm[f] = e^{-i 2pi (u+v)/64}.  Also per-block partial of sum |W|^2
// (fixed-order tree; no float atomics -> deterministic).
// ---------------------------------------------------------------------------
__global__ __launch_bounds__(256) void wfft_kernel(
    const float* __restrict__ w, float2* __restrict__ W,
    float* __restrict__ partials) {
    const int f = blockIdx.x, t = threadIdx.x;
    const int u = f / 33, v = f - u * 33;
    const float au = (2.0f * PI_F / 64.0f) * (float)u;
    const float av = (2.0f * PI_F / 64.0f) * (float)v;
    const float2 eu = make_float2(cosf(au), sinf(au));   // e^{+i2pi u/64}
    const float2 ev = make_float2(cosf(av), sinf(av));   // e^{+i2pi v/64}
    float2 Ep[3], Eq[3];
    Ep[0] = make_float2(1.0f, 0.0f); Ep[1] = eu; Ep[2] = cmul(eu, eu);
    Eq[0] = make_float2(1.0f, 0.0f); Eq[1] = ev; Eq[2] = cmul(ev, ev);
    const float2 smc = cmul(eu, ev);
    const float2 smf = make_float2(smc.x, -smc.y);       // e^{-i2pi(u+v)/64}
    float2 E[9];
    for (int p = 0; p < 3; ++p)
        for (int q = 0; q < 3; ++q)
            E[p * 3 + q] = cmul(smf, cmul(Ep[p], Eq[q]));

    float acc = 0.0f;
    float2* Wf = W + (size_t)f * 16384;
    for (int e = t; e < 16384; e += 256) {               // e = co*128 + ci
        const float* wp = w + (size_t)e * 9;
        float sr = 0.0f, si = 0.0f;
        for (int pq = 0; pq < 9; ++pq) {
            float wv = wp[pq];
            sr += wv * E[pq].x;
            si += wv * E[pq].y;
        }
        Wf[e] = make_float2(sr, si);
        acc += sr * sr + si * si;
    }
    __shared__ float red[256];
    red[t] = acc;
    __syncthreads();
    for (int s = 128; s > 0; s >>= 1) {
        if (t < s) red[t] += red[t + s];
        __syncthreads();
    }
    if (t == 0) partials[f] = red[0];
}

__global__ __launch_bounds__(256) void reduce_norm_kernel(
    const float* __restrict__ partials, float* __restrict__ sumsq) {
    __shared__ float red[256];
    const int t = threadIdx.x;
    float a = 0.0f;
    for (int i = t; i < 2112; i += 256) a += partials[i];
    red[t] = a;
    __syncthreads();
    for (int s = 128; s > 0; s >>= 1) {
        if (t < s) red[t] += red[t + s];
        __syncthreads();
    }
    if (t == 0) sumsq[0] = red[0];
}

// ---------------------------------------------------------------------------
// K4: per-frequency Cayley transform.
//   Wn = (alpha/||wfft||) * W ; A = Wn - Wn^H (skew-Hermitian)
//   M = I + A ; invert in-place (Gauss-Jordan, no pivoting: I+A is
//   guaranteed nonsingular since A is skew-Hermitian)
//   Q = Minv * (I - A) = Minv - Minv*A.
// Q is written IN PLACE over W in a WMMA-fragment-packed layout:
//   float4 Qpk[row][k2] = { Qr[row][2k2], Qr[row][2k2+1],
//                           Qi[row][2k2], Qi[row][2k2+1] }   (k2 = 0..63)
// so K5 gets both v2f WMMA A-fragments from one b128 load, no repacking.
// In-place is safe: W is fully staged into LDS (A) before any write.
// LDS: A + M = 262 KB dynamic shared (< 320 KB WGP LDS); 8 wave32 / freq.
// ---------------------------------------------------------------------------
__global__ __launch_bounds__(256) void cayley_kernel(
    float2* __restrict__ W, const float* __restrict__ sumsq,
    const float* __restrict__ alpha) {
    extern __shared__ float2 lds[];
    float2* A = lds;             // 16384 float2
    float2* M = lds + 16384;     // 16384 float2
    float2* pv = lds + 32768;    // 1 float2 (pivot broadcast)
    const int f = blockIdx.x, t = threadIdx.x;
    const float scale = alpha[0] * rsqrtf(sumsq[0]);
    float2* Wf = W + (size_t)f * 16384;

    for (int e = t; e < 16384; e += 256) {
        int i = e >> 7, j = e & 127;
        float2 wij = Wf[e];
        float2 wji = Wf[j * 128 + i];
        float2 a = make_float2(scale * (wij.x - wji.x), scale * (wij.y + wji.y));
        A[e] = a;
        M[e] = make_float2(a.x + ((i == j) ? 1.0f : 0.0f), a.y);
    }
    __syncthreads();

    // In-place Gauss-Jordan inversion of M (no pivoting).
    const int i = t >> 1;
    const int j0 = (t & 1) * 64;
    for (int k = 0; k < 128; ++k) {
        if (t == 0) {
            float2 p = M[k * 128 + k];
            float inv = 1.0f / (p.x * p.x + p.y * p.y);
            *pv = make_float2(p.x * inv, -p.y * inv);    // 1/p
        }
        __syncthreads();
        const float2 ip = *pv;
        if (t < 128) {
            int j = t;
            float2 mk = M[k * 128 + j];
            M[k * 128 + j] = (j == k) ? ip : cmul(mk, ip);
        }
        __syncthreads();
        const float2 fac = M[i * 128 + k];   // row-i multiplier (pre-elim value)
        __syncthreads();
        if (i != k) {
            for (int jj = 0; jj < 64; ++jj) {
                int j = j0 + jj;
                float2 prod = cmul(fac, M[k * 128 + j]);
                float2 mij = M[i * 128 + j];
                M[i * 128 + j] = (j == k)
                    ? make_float2(-prod.x, -prod.y)
                    : make_float2(mij.x - prod.x, mij.y - prod.y);
            }
        }
        __syncthreads();
    }

    // Q = Minv - Minv * A, two columns fused per step; packed float4 output.
    float4* Qpk = (float4*)Wf;               // [i][k2] : 128 x 64 float4
    for (int jj = 0; jj < 64; jj += 2) {
        int j = j0 + jj;
        float2 a0 = M[i * 128 + j];
        float2 a1 = M[i * 128 + j + 1];
        for (int k = 0; k < 128; ++k) {
            float2 mik = M[i * 128 + k];
            float2 p0 = cmul(mik, A[k * 128 + j]);
            float2 p1 = cmul(mik, A[k * 128 + j + 1]);
            a0.x -= p0.x; a0.y -= p0.y;
            a1.x -= p1.x; a1.y -= p1.y;
        }
        Qpk[i * 64 + (j >> 1)] = make_float4(a0.x, a1.x, a0.y, a1.y);
    }
}

// ---------------------------------------------------------------------------
// K5: yfft[f] = Q[f] @ xfft[f]  — batched complex (128x128)x(128x32) GEMM
// on V_WMMA_F32_16X16X4_F32 (full fp32 matrix engine).
//   Yr = Qr*Xr - Qi*Xi ; Yi = Qr*Xi + Qi*Xr   (4 real WMMAs / tile step;
//   -Qi folded into the A fragment: one v_xor pair, F32 WMMA has no A-neg).
// Q arrives pre-packed (float4 {qr0,qr1,qi0,qi1} per (row,k2)) -> one
// global b128 per A-fragment pair, operands land even-aligned, no movs.
// X staged into LDS in the same packed form -> one ds b128 per B-frag pair.
// One workgroup per frequency; 8 wave32 = 8 M-tiles; 2 N-tiles per wave.
// A-frag (16x4 f32): lane l: m=l&15, k-pair = 2*(l>=16).  B mirrored.
// D (16x16): vgpr r -> row r (lanes 0-15) / r+8 (lanes 16-31).
// Writes Y in place over xfft (all X reads come from LDS after barrier).
// ---------------------------------------------------------------------------
__global__ __launch_bounds__(256) void qx_wmma_kernel(
    const float4* __restrict__ Qpk, float2* __restrict__ X) {
    __shared__ float4 Xs[64 * 32];   // [k2][col] : {xr0, xr1, xi0, xi1}
    const int f = blockIdx.x, t = threadIdx.x;
    float2* Xf = X + (size_t)f * 4096;
    for (int e = t; e < 2048; e += 256) {
        int k2 = e >> 5, col = e & 31;
        float2 xa = Xf[(2 * k2) * 32 + col];
        float2 xb = Xf[(2 * k2 + 1) * 32 + col];
        Xs[e] = make_float4(xa.x, xb.x, xa.y, xb.y);
    }
    __syncthreads();

    const int wave = t >> 5, lane = t & 31;
    const int lr = lane & 15, hi = lane >> 4;
    const int mtile = wave;                       // M rows [16*wave, 16*wave+16)
    const float4* Qrow = Qpk + (size_t)f * 8192 + (size_t)(mtile * 16 + lr) * 64;
    v8f yr0 = {}, yr1 = {}, yi0 = {}, yi1 = {};

    for (int kk = 0; kk < 32; ++kk) {
        const int k2 = kk * 2 + hi;
        float4 qv = Qrow[k2];
        v2f qr  = {qv.x, qv.y};
        v2f qi  = {qv.z, qv.w};
        v2f qin = {-qv.z, -qv.w};

        float4 x0 = Xs[k2 * 32 + lr];             // N-tile 0
        float4 x1 = Xs[k2 * 32 + 16 + lr];        // N-tile 1
        v2f xr0 = {x0.x, x0.y}, xi0 = {x0.z, x0.w};
        v2f xr1 = {x1.x, x1.y}, xi1 = {x1.z, x1.w};

        yr0 = __builtin_amdgcn_wmma_f32_16x16x4_f32(false, qr,  false, xr0, (short)0, yr0, false, false);
        yr0 = __builtin_amdgcn_wmma_f32_16x16x4_f32(false, qin, false, xi0, (short)0, yr0, false, false);
        yi0 = __builtin_amdgcn_wmma_f32_16x16x4_f32(false, qr,  false, xi0, (short)0, yi0, false, false);
        yi0 = __builtin_amdgcn_wmma_f32_16x16x4_f32(false, qi,  false, xr0, (short)0, yi0, false, false);
        yr1 = __builtin_amdgcn_wmma_f32_16x16x4_f32(false, qr,  false, xr1, (short)0, yr1, false, false);
        yr1 = __builtin_amdgcn_wmma_f32_16x16x4_f32(false, qin, false, xi1, (short)0, yr1, false, false);
        yi1 = __builtin_amdgcn_wmma_f32_16x16x4_f32(false, qr,  false, xi1, (short)0, yi1, false, false);
        yi1 = __builtin_amdgcn_wmma_f32_16x16x4_f32(false, qi,  false, xr1, (short)0, yi1, false, false);
    }

    for (int r = 0; r < 8; ++r) {
        int row = mtile * 16 + hi * 8 + r;
        Xf[row * 32 + lr]      = make_float2(yr0[r], yi0[r]);
        Xf[row * 32 + 16 + lr] = make_float2(yr1[r], yi1[r]);
    }
}

// ---------------------------------------------------------------------------
// K6: irfft2 + bias. Per (b,co) image: gather Y[u][v], inverse column DFT
// (with 1/64), then real inverse row transform (Hermitian completion, 1/64),
// add bias, write y (b, co, 64, 64).
// ---------------------------------------------------------------------------
__global__ __launch_bounds__(256) void irfft2_bias_kernel(
    const float2* __restrict__ Y, const float* __restrict__ bias,
    float* __restrict__ out) {
    __shared__ float2 Yl[64 * 33];
    __shared__ float2 T[64 * 33];
    __shared__ float ct[64], st[64];
    const int blk = blockIdx.x, t = threadIdx.x;
    const int b = blk >> 7, co = blk & 127;
    for (int e = t; e < 2112; e += 256)
        Yl[e] = Y[(size_t)e * 4096 + co * 32 + b];
    if (t < 64) {
        float ang = (2.0f * PI_F / 64.0f) * (float)t;
        ct[t] = cosf(ang);
        st[t] = sinf(ang);
    }
    __syncthreads();
    // stage 1: T[nrow][v] = (1/64) sum_u Yl[u][v] e^{+i 2pi u nrow / 64}
    for (int o = t; o < 2112; o += 256) {
        int nrow = o / 33, v = o - nrow * 33;
        float re = 0.0f, im = 0.0f;
        for (int u = 0; u < 64; ++u) {
            int idx = (u * nrow) & 63;
            float2 s = Yl[u * 33 + v];
            float cc = ct[idx], ss = st[idx];
            re += s.x * cc - s.y * ss;
            im += s.x * ss + s.y * cc;
        }
        T[o] = make_float2(re * (1.0f / 64.0f), im * (1.0f / 64.0f));
    }
    __syncthreads();
    const float bv = bias[co];
    // stage 2: y[nrow][ncol] = (1/64)[T0.re + (-1)^ncol T32.re
    //                                + 2 sum_{v=1}^{31}(Tr cos - Ti sin)]
    for (int o = t; o < 4096; o += 256) {
        int nrow = o >> 6, ncol = o & 63;
        const float2* Tp = T + nrow * 33;
        float acc = Tp[0].x + ((ncol & 1) ? -Tp[32].x : Tp[32].x);
        float s2 = 0.0f;
        for (int v = 1; v < 32; ++v) {
            int idx = (v * ncol) & 63;
            s2 += Tp[v].x * ct[idx] - Tp[v].y * st[idx];
        }
        acc += 2.0f * s2;
        out[(size_t)blk * 4096 + o] = acc * (1.0f / 64.0f) + bv;
    }
}

// ---------------------------------------------------------------------------
// Host launcher. Workspace layout (all recomputed every call; deterministic):
//   ws_rows : 4096*2112 c64  (69.2 MB)   row-rfft intermediate
//   ws_xfft : 2112*4096 c64  (69.2 MB)   xfft, overwritten in place by yfft
//   ws_W    : 2112*16384 c64 (276.8 MB)  wfft, overwritten in place by Q(packed)
//   ws_part : 2112 f32 ; ws_sumsq : 1 f32
// ---------------------------------------------------------------------------
extern "C" void kernel_launch(void* const* d_in, const int* in_sizes, int n_in,
                              void* d_out, int out_size, void* d_ws,
                              size_t ws_size, hipStream_t stream) {
    const float* x      = (const float*)d_in[0];   // (32,128,64,64)
    const float* weight = (const float*)d_in[1];   // (128,128,3,3)
    const float* bias   = (const float*)d_in[2];   // (128,)
    const float* alpha  = (const float*)d_in[3];   // (1,)
    float* out = (float*)d_out;

    char* ws = (char*)d_ws;
    size_t off = 0;
    float2* ws_rows = (float2*)(ws + off); off += (size_t)4096 * 2112 * sizeof(float2);
    float2* ws_xfft = (float2*)(ws + off); off += (size_t)2112 * 4096 * sizeof(float2);
    float2* ws_W    = (float2*)(ws + off); off += (size_t)2112 * 16384 * sizeof(float2);
    float*  ws_part = (float*)(ws + off);  off += (size_t)2112 * sizeof(float);
    float*  ws_sumsq= (float*)(ws + off);  off += 256;

    rfft_rows_kernel  <<<4096, 256, 0, stream>>>(x, ws_rows);
    fft_cols_kernel   <<<4096, 256, 0, stream>>>(ws_rows, ws_xfft);
    wfft_kernel       <<<2112, 256, 0, stream>>>(weight, ws_W, ws_part);
    reduce_norm_kernel<<<1,    256, 0, stream>>>(ws_part, ws_sumsq);
    // dynamic LDS: A + M + pivot = (16384 + 16384 + 8) float2 = 262,208 B
    cayley_kernel     <<<2112, 256, (32768 + 8) * sizeof(float2), stream>>>(
        ws_W, ws_sumsq, alpha);
    qx_wmma_kernel    <<<2112, 256, 0, stream>>>((const float4*)ws_W, ws_xfft);
    irfft2_bias_kernel<<<4096, 256, 0, stream>>>(ws_xfft, bias, out);
}